// BuildCostVolume_11630771437833
// MI455X (gfx1250) — compile-verified
//
#include <hip/hip_runtime.h>

typedef float v2f __attribute__((ext_vector_type(2)));
typedef float v4f __attribute__((ext_vector_type(4)));
typedef float v8f __attribute__((ext_vector_type(8)));

#define BB 4
#define CC 64
#define HH 256
#define WW 512
#define DD 96
#define BAND 112   // right-band width: 96 disparities + 16 tile columns
#define NT 7       // gram tiles per 16-wide w-tile

// LDS layouts (channel-pair packed):
//   Lp[p][w]  = { left [2p][w], left [2p+1][w] }   p in [0,32), w in [0,16)
//   Rp[p][q]  = { right[2p][q], right[2p+1][q] }   p in [0,32), q in [0,112)
// Inner loop: pair index p = 2*ks + hi  ->  one ds_load_b64 per fragment,
// immediate offsets ks*256B (A) / ks*1792B (B), data lands in an even VGPR pair.

__launch_bounds__(256)
__global__ void cost_volume_wmma_f32(const float* __restrict__ left,
                                     const float* __restrict__ right,
                                     const float* __restrict__ shift,
                                     float* __restrict__ out)
{
    __shared__ float Lp[32 * 16 * 2];     //  4 KB
    __shared__ float Rp[32 * BAND * 2];   // 28 KB
    __shared__ float Gs[16 * BAND];       //  7 KB gram band: Gs[wl][q]
    __shared__ float Ss[DD];

    const int tid = threadIdx.x;
    const int wt  = blockIdx.x;           // w-tile index 0..31
    const int h   = blockIdx.y;
    const int b   = blockIdx.z;
    const int w0  = wt * 16;
    const int wq0 = w0 - 96;              // leftmost band column (may be <0)

    // ---- stage left tile (packed pairs): 32 p x 4 quads = 128 units
    if (tid < 128) {
        int p  = tid >> 2;                // channel pair 0..31
        int wq = (tid & 3) * 4;           // 0,4,8,12
        const float* p0 = left + (((size_t)b * CC + 2 * p    ) * HH + h) * WW + w0 + wq;
        const float* p1 = left + (((size_t)b * CC + 2 * p + 1) * HH + h) * WW + w0 + wq;
        v4f a0 = *(const v4f*)p0;
        v4f a1 = *(const v4f*)p1;
        #pragma unroll
        for (int j = 0; j < 4; ++j) {
            v2f pr = {a0[j], a1[j]};
            *(v2f*)&Lp[(p * 16 + wq + j) * 2] = pr;
        }
    }
    // ---- stage right band (packed pairs): 32 p x 28 quads = 896 units
    #pragma unroll
    for (int k = 0; k < 4; ++k) {
        int idx = tid + k * 256;
        if (idx < 896) {
            int p  = idx & 31;            // channel pair
            int qd = idx >> 5;            // quad 0..27
            int wg = wq0 + qd * 4;        // multiple of 4; <0 -> all 4 elems OOB
            v4f r0 = {}, r1 = {};
            if (wg >= 0) {
                const float* p0 = right + (((size_t)b * CC + 2 * p    ) * HH + h) * WW + wg;
                const float* p1 = right + (((size_t)b * CC + 2 * p + 1) * HH + h) * WW + wg;
                r0 = *(const v4f*)p0;
                r1 = *(const v4f*)p1;
            }
            #pragma unroll
            for (int j = 0; j < 4; ++j) {
                v2f pr = {r0[j], r1[j]};
                *(v2f*)&Rp[(p * BAND + qd * 4 + j) * 2] = pr;
            }
        }
    }
    if (tid < DD) Ss[tid] = shift[tid];
    __syncthreads();

    // ---- Gram tiles via V_WMMA_F32_16X16X4_F32: wave t computes 16x16 tile t
    const int wave = tid >> 5;
    const int lane = tid & 31;
    if (wave < NT) {                      // wave-uniform branch: EXEC all ones
        const int col = lane & 15;
        const int hi  = lane >> 4;        // 0 (K=c0,c0+1) / 1 (K=c0+2,c0+3)
        const int qb  = 96 - 16 * wave;   // band column of tile's first wr
        const float* Ab = &Lp[(hi * 16 + col) * 2];
        const float* Bb = &Rp[(hi * BAND + qb + col) * 2];
        v8f acc = {};
        #pragma unroll
        for (int ks = 0; ks < 16; ++ks) { // K = 64 in slices of 4
            v2f a  = *(const v2f*)(Ab + ks * 64);           // +256 B / step
            v2f bf = *(const v2f*)(Bb + ks * (2 * BAND * 2)); // +1792 B / step
            acc = __builtin_amdgcn_wmma_f32_16x16x4_f32(
                      false, a, false, bf, (short)0, acc, false, false);
        }
        // D layout: VGPR i -> M = i + 8*(lane>=16), N = col
        const int mhi = hi * 8;
        #pragma unroll
        for (int i = 0; i < 8; ++i)
            Gs[(mhi + i) * BAND + qb + col] = acc[i];
    }
    __syncthreads();

    // ---- epilogue: per-disparity fractional blend of gram diagonals
    #pragma unroll
    for (int k = 0; k < 6; ++k) {
        int e  = tid + k * 256;           // 96*16 = 1536 outputs per block
        int d  = e >> 4;
        int wl = e & 15;
        float s    = Ss[d];
        int   i0   = (int)floorf(s);
        float frac = s - (float)i0;
        int q0  = wl - i0 + 96;           // band column for integer shift i0
        int wrg = w0 + wl - i0;           // global source column
        float g0 = (wrg >= 0 && q0 >= 0 && q0 <  BAND) ? Gs[wl * BAND + q0]     : 0.f;
        float g1 = (wrg >= 1 && q0 >= 1 && q0 <= BAND) ? Gs[wl * BAND + q0 - 1] : 0.f;
        out[(((size_t)b * DD + d) * HH + h) * WW + w0 + wl] =
            (1.f - frac) * g0 + frac * g1;
    }
}

extern "C" void kernel_launch(void* const* d_in, const int* in_sizes, int n_in,
                              void* d_out, int out_size, void* d_ws, size_t ws_size,
                              hipStream_t stream) {
    const float* left  = (const float*)d_in[0];
    const float* right = (const float*)d_in[1];
    const float* shift = (const float*)d_in[2];
    float* out = (float*)d_out;
    dim3 grid(WW / 16, HH, BB);   // 32 x 256 x 4 = 32768 workgroups
    cost_volume_wmma_f32<<<grid, 256, 0, stream>>>(left, right, shift, out);
}